// gtopk_62826781606157
// MI455X (gfx1250) — compile-verified
//
#include <hip/hip_runtime.h>

typedef float v2f __attribute__((ext_vector_type(2)));
typedef float v8f __attribute__((ext_vector_type(8)));

#define GS      128
#define PARAM   32
#define KNEI    8
#define NUM_C   10000.0f
#define EMPTYV  100000000.0f
#define SELF_C  100.0f
#define DSTR    129   // padded LDS row stride (bank-conflict-free row scans)

__global__ __launch_bounds__(256)
void gtopk_62826781606157_kernel(const float* __restrict__ x,
                                 const float* __restrict__ metrik,
                                 float* __restrict__ out_rel,
                                 float* __restrict__ out_x) {
  __shared__ float sm[PARAM];         // metrik
  __shared__ float xs[GS * PARAM];    // x[b]           (B operand, row-major)
  __shared__ float asc[GS * PARAM];   // -2*m*x[b]      (A operand, row-major)
  __shared__ float dmat[GS * DSTR];   // distance matrix (padded rows)
  __shared__ float srow[GS];          // s_i = sum_p m_p x_ip^2
  __shared__ float flv[GS];           // flag column (feature 7)
  __shared__ float kthv[GS];          // 8th-smallest per row
  __shared__ float facv[GS];          // 8/(numnei+eps)

  const int b = blockIdx.x;
  const int t = threadIdx.x;
  const float* xb = x + (size_t)b * GS * PARAM;

  if (t < PARAM) sm[t] = metrik[t];
  __syncthreads();

  // Stage x and scaled-A into LDS
  for (int idx = t; idx < GS * PARAM; idx += 256) {
    float v = xb[idx];
    xs[idx]  = v;
    asc[idx] = -2.0f * sm[idx & (PARAM - 1)] * v;
  }
  __syncthreads();

  if (t < GS) {
    float s = 0.0f;
#pragma unroll
    for (int p = 0; p < PARAM; ++p) {
      float v = xs[t * PARAM + p];
      s = fmaf(sm[p] * v, v, s);
    }
    srow[t] = s;
    flv[t]  = xs[t * PARAM + 7];
  }
  __syncthreads();

  // ---- WMMA phase: dmat tile = -2 * (m.x) @ x^T via V_WMMA_F32_16X16X4_F32 ----
  // A 16x4 f32 fragment: lanes 0-15 -> M=lane, VGPR0/1 = K=0/1; lanes 16-31 -> K=2/3.
  // B 4x16 f32 fragment: symmetric (lane = N), B[k][n] = xs[n][k].
  const int wave = t >> 5;
  const int lane = t & 31;
  const int n    = lane & 15;
  const int kh   = lane >> 4;        // 0: K pair {0,1}, 1: K pair {2,3}
  const int rowA = (wave << 4) + n;

  v2f afrag[8];
#pragma unroll
  for (int k = 0; k < 8; ++k)
    afrag[k] = *(const v2f*)&asc[rowA * PARAM + 4 * k + 2 * kh];

  for (int c = 0; c < 8; ++c) {
    v8f acc = {};
    const int colB = (c << 4) + n;
#pragma unroll
    for (int k = 0; k < 8; ++k) {
      v2f bfrag = *(const v2f*)&xs[colB * PARAM + 4 * k + 2 * kh];
      acc = __builtin_amdgcn_wmma_f32_16x16x4_f32(
          /*neg_a=*/false, afrag[k], /*neg_b=*/false, bfrag,
          /*c_mod=*/(short)0, acc, /*reuse_a=*/false, /*reuse_b=*/false);
    }
    // C/D layout: VGPR v, lanes 0-15 -> M=v; lanes 16-31 -> M=v+8. N = lane&15.
#pragma unroll
    for (int v = 0; v < 8; ++v) {
      int row = (wave << 4) + v + (kh << 3);
      dmat[row * DSTR + (c << 4) + n] = acc[v];
    }
  }
  __syncthreads();

  // ---- Fixup: d = -2G + s_i + s_j + (1 - fl_i*fl_j)*EMPTY + SELF_C*eye ----
  for (int idx = t; idx < GS * GS; idx += 256) {
    int i = idx >> 7, j = idx & (GS - 1);
    float d = dmat[i * DSTR + j] + srow[i] + srow[j]
              + (1.0f - flv[i] * flv[j]) * EMPTYV;
    if (i == j) d += SELF_C;
    dmat[i * DSTR + j] = d;
  }
  __syncthreads();

  // ---- Per-row 8th-smallest (branchless sorted-insert network) + numnei ----
  if (t < GS) {
    const float* row = &dmat[t * DSTR];
    float b0 = 3.0e38f, b1 = 3.0e38f, b2 = 3.0e38f, b3 = 3.0e38f;
    float b4 = 3.0e38f, b5 = 3.0e38f, b6 = 3.0e38f, b7 = 3.0e38f;
    for (int j = 0; j < GS; ++j) {
      float v = row[j], lo;
      lo = fminf(b0, v); v = fmaxf(b0, v); b0 = lo;
      lo = fminf(b1, v); v = fmaxf(b1, v); b1 = lo;
      lo = fminf(b2, v); v = fmaxf(b2, v); b2 = lo;
      lo = fminf(b3, v); v = fmaxf(b3, v); b3 = lo;
      lo = fminf(b4, v); v = fmaxf(b4, v); b4 = lo;
      lo = fminf(b5, v); v = fmaxf(b5, v); b5 = lo;
      lo = fminf(b6, v); v = fmaxf(b6, v); b6 = lo;
      lo = fminf(b7, v); b7 = lo;
    }
    float kd = b7;                       // 8th smallest; vb = -kd
    float fi = flv[t];
    float num = 0.0f;
    for (int j = 0; j < GS; ++j) {
      float su = row[j] - kd;            // d + vb
      float r  = fminf(fmaxf(fmaf(-NUM_C, su, 1.0f), 0.0f), 1.0f);
      num = fmaf(r, fi * flv[j], num);
    }
    kthv[t] = kd;
    facv[t] = (float)KNEI / (num + 1e-11f);
  }
  __syncthreads();

  // ---- Coalesced output: rel = clamp(1 - C*(d-kth),0,1) * isval * factor ----
  float* orel = out_rel + (size_t)b * GS * GS;
  for (int idx = t; idx < GS * GS; idx += 256) {
    int i = idx >> 7, j = idx & (GS - 1);
    float su = dmat[i * DSTR + j] - kthv[i];
    float r  = fminf(fmaxf(fmaf(-NUM_C, su, 1.0f), 0.0f), 1.0f);
    orel[idx] = r * flv[i] * flv[j] * facv[i];
  }
  // ---- Second tuple output: x passthrough ----
  float* ox = out_x + (size_t)b * GS * PARAM;
  for (int idx = t; idx < GS * PARAM; idx += 256) ox[idx] = xs[idx];
}

extern "C" void kernel_launch(void* const* d_in, const int* in_sizes, int n_in,
                              void* d_out, int out_size, void* d_ws, size_t ws_size,
                              hipStream_t stream) {
  const float* x      = (const float*)d_in[0];   // 256*128*32
  const float* metrik = (const float*)d_in[1];   // 32
  float* out_rel = (float*)d_out;                          // 256*128*128
  float* out_x   = (float*)d_out + (size_t)256 * GS * GS;  // 256*128*32
  gtopk_62826781606157_kernel<<<256, 256, 0, stream>>>(x, metrik, out_rel, out_x);
}